// Pool_22076131901972
// MI455X (gfx1250) — compile-verified
//
#include <hip/hip_runtime.h>

// ---------------------------------------------------------------------------
// Single-query attention block, algebraically reduced to be bandwidth-bound.
//   scores = features @ (Wk_head @ q_head)   [S x H]   (WMMA f32 16x16x4)
//   A      = softmax(scores / sqrt(P))                  (per-head)
//   ctx    = A^T-weighted sum of features    [H x E]   (WMMA f32 16x16x4)
//   O      = LN( q + ctx @ Wv + bv ) ; O += relu(O@Wo+bo) ; LN
// Dominant traffic: features read twice (2 x 256 MB) ~= 22 us @ 23.3 TB/s.
// ---------------------------------------------------------------------------

typedef __attribute__((ext_vector_type(2))) float v2f;
typedef __attribute__((ext_vector_type(8))) float v8f;

#define S_      32768
#define E_      2048
#define P_      2048
#define H_      16
#define DS_     128
#define LN_EPS  1e-5f
#define SEGS    32
#define SEGLEN  1024          // S_ / SEGS
#define ECHUNKS 128           // E_ / 16

// ---- stage 0: wkq[e,h] = sum_d Wk[e, h*128+d] * q[h*128+d]  (tiny) --------
__global__ void k_wkq(const float* __restrict__ Wk, const float* __restrict__ q,
                      float* __restrict__ wkq) {
    int idx = blockIdx.x * blockDim.x + threadIdx.x;   // = e*16 + h
    int e = idx >> 4, h = idx & 15;
    const float* wrow = Wk + (size_t)e * P_ + h * DS_;
    const float* qh   = q + h * DS_;
    float acc = 0.f;
#pragma unroll 8
    for (int d = 0; d < DS_; ++d) acc += wrow[d] * qh[d];
    wkq[idx] = acc;
}

// ---- stage 1: scores[s,h] = features[s,:] . wkq[:,h]  via WMMA f32 --------
// One wave per 16-row tile of S.  A = feat tile (16x4), B = wkq chunk (4x16).
__global__ void k_scores(const float* __restrict__ feat,
                         const float* __restrict__ wkq,
                         float* __restrict__ scores) {
    const int wave = threadIdx.x >> 5;
    const int lane = threadIdx.x & 31;
    const int s0   = (blockIdx.x * 8 + wave) * 16;
    const int hf   = lane >> 4;          // half-wave: K offset 0 or 2
    const int lm   = lane & 15;          // A: row M ; B/C: column N
    const float* arow  = feat + (size_t)(s0 + lm) * E_ + hf * 2;
    const float* bbase = wkq + hf * 2 * H_ + lm;
    v8f c = {};
#pragma unroll 4
    for (int k0 = 0; k0 < E_; k0 += 4) {
        v2f a = *(const v2f*)(arow + k0);          // K=k0+hf*2, k0+hf*2+1
        v2f b;
        b.x = bbase[k0 * H_];                      // wkq[k0+hf*2    , lm]
        b.y = bbase[k0 * H_ + H_];                 // wkq[k0+hf*2 + 1, lm]
        c = __builtin_amdgcn_wmma_f32_16x16x4_f32(false, a, false, b,
                                                  (short)0, c, false, false);
    }
    const float scale = 0.022097086912079608f;     // 1/sqrt(P)
    float* out = scores + (size_t)(s0 + hf * 8) * H_ + lm;
#pragma unroll
    for (int r = 0; r < 8; ++r) out[r * H_] = c[r] * scale;
}

// ---- stage 2: softmax over S per head (in place) --------------------------
__global__ void k_softmax(float* __restrict__ scores) {
    const int h = blockIdx.x;
    __shared__ float red[256];
    float m = -3.4e38f;
    for (int s = threadIdx.x; s < S_; s += 256) m = fmaxf(m, scores[s * H_ + h]);
    red[threadIdx.x] = m; __syncthreads();
    for (int o = 128; o > 0; o >>= 1) {
        if (threadIdx.x < o) red[threadIdx.x] = fmaxf(red[threadIdx.x], red[threadIdx.x + o]);
        __syncthreads();
    }
    m = red[0]; __syncthreads();
    float sum = 0.f;
    for (int s = threadIdx.x; s < S_; s += 256) {
        float ev = __expf(scores[s * H_ + h] - m);
        scores[s * H_ + h] = ev; sum += ev;
    }
    red[threadIdx.x] = sum; __syncthreads();
    for (int o = 128; o > 0; o >>= 1) {
        if (threadIdx.x < o) red[threadIdx.x] += red[threadIdx.x + o];
        __syncthreads();
    }
    float inv = 1.f / red[0];
    for (int s = threadIdx.x; s < S_; s += 256) scores[s * H_ + h] *= inv;
}

// ---- stage 3: ctx partials: part[seg,ec,h,n] = sum_{s in seg} A[s,h]*feat[s,e0+n]
// A = A_sm^T (16 heads x 4 s) from LDS, B = feat (4 s x 16 e).  Wave per e-chunk.
__global__ void k_ctx_partial(const float* __restrict__ feat,
                              const float* __restrict__ Asm,
                              float* __restrict__ part) {
    __shared__ float lA[SEGLEN * H_];               // 64 KB of the 320 KB LDS
    const int seg  = blockIdx.x >> 4;
    const int grp  = blockIdx.x & 15;
    const int wave = threadIdx.x >> 5;
    const int lane = threadIdx.x & 31;
    const int ec   = grp * 8 + wave;
    const int e0   = ec * 16;
    const int sbase = seg * SEGLEN;
    for (int i = threadIdx.x; i < SEGLEN * H_; i += 256)
        lA[i] = Asm[(size_t)sbase * H_ + i];
    __syncthreads();
    const int hf = lane >> 4, lm = lane & 15;
    const float* bbase = feat + (size_t)(sbase + hf * 2) * E_ + e0 + lm;
    v8f c = {};
#pragma unroll 4
    for (int s0 = 0; s0 < SEGLEN; s0 += 4) {
        v2f a, b;
        a.x = lA[(s0 + hf * 2) * H_ + lm];          // A[M=head lm][K=s0+hf*2]
        a.y = lA[(s0 + hf * 2 + 1) * H_ + lm];
        b.x = bbase[(size_t)s0 * E_];               // feat[K row][N=e0+lm]
        b.y = bbase[(size_t)(s0 + 1) * E_];
        c = __builtin_amdgcn_wmma_f32_16x16x4_f32(false, a, false, b,
                                                  (short)0, c, false, false);
    }
    float* out = part + ((size_t)(seg * ECHUNKS + ec) * 16 + hf * 8) * 16 + lm;
#pragma unroll
    for (int r = 0; r < 8; ++r) out[r * 16] = c[r];
}

// ---- stage 4: ctx[h,e] = sum_seg part  (fixed order -> deterministic) -----
__global__ void k_ctx_reduce(const float* __restrict__ part, float* __restrict__ ctx) {
    int idx = blockIdx.x * 256 + threadIdx.x;       // h*E + e
    int h = idx >> 11, e = idx & (E_ - 1);
    int ec = e >> 4, n = e & 15;
    float acc = 0.f;
#pragma unroll 8
    for (int seg = 0; seg < SEGS; ++seg)
        acc += part[((size_t)(seg * ECHUNKS + ec) * 16 + h) * 16 + n];
    ctx[idx] = acc;
}

// ---- stage 5: head_out[j] = q[j] + ctx[h,:] . Wv[:,j] + bv[j] -------------
__global__ void k_headout(const float* __restrict__ ctx, const float* __restrict__ Wv,
                          const float* __restrict__ q, const float* __restrict__ bv,
                          float* __restrict__ O0) {
    int j = blockIdx.x * 256 + threadIdx.x;         // 0..P-1
    int h = j >> 7;
    const float* cr = ctx + (size_t)h * E_;
    float acc = 0.f;
#pragma unroll 4
    for (int e = 0; e < E_; ++e) acc += cr[e] * Wv[(size_t)e * P_ + j];
    O0[j] = q[j] + acc + bv[j];
}

// ---- layer norm over the 2048-vector (single block) -----------------------
__global__ void k_layernorm(const float* __restrict__ x, const float* __restrict__ g,
                            const float* __restrict__ b, float* __restrict__ y) {
    __shared__ float red[256];
    float s = 0.f;
    for (int i = threadIdx.x; i < P_; i += 256) s += x[i];
    red[threadIdx.x] = s; __syncthreads();
    for (int o = 128; o > 0; o >>= 1) {
        if (threadIdx.x < o) red[threadIdx.x] += red[threadIdx.x + o];
        __syncthreads();
    }
    float mean = red[0] / P_; __syncthreads();
    float v = 0.f;
    for (int i = threadIdx.x; i < P_; i += 256) { float d = x[i] - mean; v += d * d; }
    red[threadIdx.x] = v; __syncthreads();
    for (int o = 128; o > 0; o >>= 1) {
        if (threadIdx.x < o) red[threadIdx.x] += red[threadIdx.x + o];
        __syncthreads();
    }
    float inv = rsqrtf(red[0] / P_ + LN_EPS);
    for (int i = threadIdx.x; i < P_; i += 256)
        y[i] = (x[i] - mean) * inv * g[i] + b[i];
}

// ---- residual MLP: O2[j] = O1[j] + relu(O1 . Wo[:,j] + bo[j]) -------------
__global__ void k_mlp(const float* __restrict__ O1, const float* __restrict__ Wo,
                      const float* __restrict__ bo, float* __restrict__ O2) {
    __shared__ float xo[P_];
    for (int i = threadIdx.x; i < P_; i += 256) xo[i] = O1[i];
    __syncthreads();
    int j = blockIdx.x * 256 + threadIdx.x;
    float acc = bo[j];
#pragma unroll 4
    for (int i = 0; i < P_; ++i) acc += xo[i] * Wo[(size_t)i * P_ + j];
    O2[j] = O1[j] + fmaxf(acc, 0.f);
}

extern "C" void kernel_launch(void* const* d_in, const int* in_sizes, int n_in,
                              void* d_out, int out_size, void* d_ws, size_t ws_size,
                              hipStream_t stream) {
    const float* features = (const float*)d_in[0];
    const float* query    = (const float*)d_in[1];
    const float* Wk       = (const float*)d_in[2];
    // d_in[3] = bk: per-head constant shift of scores -> cancelled by softmax.
    const float* Wv       = (const float*)d_in[4];
    const float* bv       = (const float*)d_in[5];
    const float* Wo       = (const float*)d_in[6];
    const float* bo       = (const float*)d_in[7];
    const float* g0       = (const float*)d_in[8];
    const float* b0       = (const float*)d_in[9];
    const float* g1       = (const float*)d_in[10];
    const float* b1       = (const float*)d_in[11];
    (void)in_sizes; (void)n_in; (void)out_size; (void)ws_size;

    float* ws     = (float*)d_ws;
    float* wkq    = ws;                                   // E*H           = 32768
    float* scores = wkq    + E_ * H_;                     // S*H           = 524288
    float* part   = scores + (size_t)S_ * H_;             // SEGS*ECH*256  = 1048576
    float* ctx    = part   + (size_t)SEGS * ECHUNKS * 256;// H*E           = 32768
    float* O0     = ctx    + H_ * E_;                     // P
    float* O1     = O0 + P_;
    float* O2     = O1 + P_;

    k_wkq        <<<(E_ * H_) / 256, 256, 0, stream>>>(Wk, query, wkq);
    k_scores     <<<(S_ / 16) / 8,   256, 0, stream>>>(features, wkq, scores);
    k_softmax    <<<H_,              256, 0, stream>>>(scores);
    k_ctx_partial<<<SEGS * 16,       256, 0, stream>>>(features, scores, part);
    k_ctx_reduce <<<(H_ * E_) / 256, 256, 0, stream>>>(part, ctx);
    k_headout    <<<P_ / 256,        256, 0, stream>>>(ctx, Wv, query, bv, O0);
    k_layernorm  <<<1,               256, 0, stream>>>(O0, g0, b0, O1);
    k_mlp        <<<P_ / 256,        256, 0, stream>>>(O1, Wo, bo, O2);
    k_layernorm  <<<1,               256, 0, stream>>>(O2, g1, b1, (float*)d_out);
}